// VQVAE1_34325378630027
// MI455X (gfx1250) — compile-verified
//
#include <hip/hip_runtime.h>

typedef __attribute__((ext_vector_type(2))) float        v2f;
typedef __attribute__((ext_vector_type(8))) float        v8f;
typedef __attribute__((ext_vector_type(4))) unsigned int v4u;
typedef __attribute__((ext_vector_type(8))) int          v8i;
typedef __attribute__((ext_vector_type(4))) int          v4i;

#define PROMPT_LEN 2048
#define VOCAB      49408
#define DIM        768
#define P_BLOCKS   16                      // 128 prompt rows per block
#define V_CHUNKS   16
#define V_PER_CHUNK (VOCAB / V_CHUNKS)     // 3088
#define TILES_PER_CHUNK (V_PER_CHUNK / 16) // 193

// TDM pads LDS dest: 2 DWORDs inserted every 256 DWORDs stored.
// Row r, col k lands at DWORD offset 774*r + k + 2*(k>>8)   (row stride 774 == 6 mod 64
// -> the 16 B-matrix rows hit distinct banks; pads never split an even-offset pair).
#define ROW_DW  774
#define TILE_DW 12384                      // 16*774 (covers last element + slack)

// tensor_load_to_lds builtin arity differs between toolchains; the TDM header
// ships only with the 6-arg toolchain.
#if __has_include(<hip/amd_detail/amd_gfx1250_TDM.h>)
#define TDM_LOAD(g0, g1, g2, g3) \
    __builtin_amdgcn_tensor_load_to_lds((g0), (g1), (g2), (g3), (v8i)0, 0)
#else
#define TDM_LOAD(g0, g1, g2, g3) \
    __builtin_amdgcn_tensor_load_to_lds((g0), (g1), (g2), (g3), 0)
#endif

// ---------------------------------------------------------------- c^2 norms
__global__ __launch_bounds__(256) void vq_csq_kernel(const float* __restrict__ clip,
                                                     float* __restrict__ csq) {
    const int wave = threadIdx.x >> 5;
    const int lane = threadIdx.x & 31;
    const int row  = blockIdx.x * 8 + wave;          // 6176 blocks * 8 == 49408
    const float* r = clip + (size_t)row * DIM;
    float s = 0.f;
#pragma unroll
    for (int j = 0; j < DIM / 32; ++j) {
        float x = r[lane + j * 32];
        s += x * x;
    }
#pragma unroll
    for (int m = 16; m >= 1; m >>= 1) s += __shfl_xor(s, m, 32);
    if (lane == 0) csq[row] = s;
}

// ------------------------------------------------- WMMA cross-term + argmin
// argmin_v (c_sq[v] - 2 * <p, c_v>)   (p^2 is constant per row -> dropped)
__global__ __launch_bounds__(256) void vq_nn_kernel(const float* __restrict__ prompt,
                                                    const float* __restrict__ clip,
                                                    const float* __restrict__ csq,
                                                    float* __restrict__ partD,
                                                    int*   __restrict__ partI) {
    __shared__ float lds[2 * TILE_DW];               // double-buffered TDM dest (~97 KB)

    const int tid  = threadIdx.x;
    const int wave = tid >> 5;
    const int lane = tid & 31;
    const int half = lane >> 4;                       // 0: K+{0,1}, 1: K+{2,3}
    const int l15  = lane & 15;                       // M for A, N for B/C/D

    const int pRowBase = blockIdx.x * 128 + wave * 16;   // this wave's 16 rows
    const int vStart   = blockIdx.y * V_PER_CHUNK;

    // A fragment source: row M = l15, K-pair chosen by lane half (ISA 16x4 f32 layout)
    const float* aRow = prompt + (size_t)(pRowBase + l15) * DIM + 2 * half;

    const unsigned ldsBase = (unsigned)(uintptr_t)(&lds[0]);

    // Issue one 2-D TDM descriptor: 16 x 768 f32 tile of clip -> LDS buffer.
    auto tdm_issue = [&](int t) {
        const int      buf   = t & 1;
        const uint64_t gaddr = (uint64_t)(uintptr_t)(clip + (size_t)(vStart + t * 16) * DIM);
        // group0: count=1 | lds_addr | global_addr[56:0] | type=2
        v4u g0 = { 1u,
                   ldsBase + (unsigned)buf * (TILE_DW * 4u),
                   (unsigned)gaddr,
                   (unsigned)(gaddr >> 32) | (2u << 30) };
        // group1: data_size=4B(2) | pad_enable | pad_interval=256DW(7) | pad_amount=2DW(1)
        //         tensor_dim0=768, tensor_dim1=VOCAB, tile_dim0=768, tile_dim1=16,
        //         tensor_dim0_stride=768
        v8i g1 = { (int)0x03D20000,
                   (int)(768u << 16),          // tensor_dim0[15:0] << 16
                   (int)(49408u << 16),        // tensor_dim1[15:0] << 16
                   (int)(768u << 16),          // tile_dim0 << 16
                   16,                         // tile_dim1
                   768,                        // tensor_dim0_stride[31:0]
                   0, 0 };
        v4i gz = { 0, 0, 0, 0 };
        TDM_LOAD(g0, g1, gz, gz);
    };

    float bestD[8];
    int   bestI[8];
#pragma unroll
    for (int v = 0; v < 8; ++v) { bestD[v] = 3.4e38f; bestI[v] = 0; }

    if (wave == 0) tdm_issue(0);                      // prime the pipeline

    for (int t = 0; t < TILES_PER_CHUNK; ++t) {
        __syncthreads();                              // buffer[(t+1)&1] free again
        if (wave == 0) {
            if (t + 1 < TILES_PER_CHUNK) {
                tdm_issue(t + 1);                     // DMA next tile (other buffer)
                __builtin_amdgcn_s_wait_tensorcnt(1); // tile t landed (in-order TDM)
            } else {
                __builtin_amdgcn_s_wait_tensorcnt(0);
            }
        }
        __syncthreads();                              // tile t visible to all waves

        // B base for this lane's N-row inside the padded tile, + this half's K pair
        const float* bRow = lds + (t & 1) * TILE_DW + ROW_DW * l15 + 2 * half;

        v8f acc = {0.f, 0.f, 0.f, 0.f, 0.f, 0.f, 0.f, 0.f};
        // 3 segments of 256 DWORDs: TDM pad offset (2 DW / 256 DW) is constant
        // inside a segment, so it folds into the segment base pointer.
        for (int seg = 0; seg < 3; ++seg) {
            const float* aSeg = aRow + seg * 256;
            const float* bSeg = bRow + seg * 258;     // 256 data + 2 pad DWORDs
#pragma unroll 4
            for (int k = 0; k < 256; k += 4) {
                v2f a = *(const v2f*)(aSeg + k);      // A: 16x4 f32 fragment
                v2f b = *(const v2f*)(bSeg + k);      // B: 4x16 f32 fragment
                acc = __builtin_amdgcn_wmma_f32_16x16x4_f32(
                    false, a, false, b, (short)0, acc, false, false);
            }
        }

        const int   n0 = vStart + t * 16;
        const int   n  = n0 + l15;                    // this lane's column id
        const float cs = csq[n];
#pragma unroll
        for (int v = 0; v < 8; ++v) {                 // M = v + 8*half, N = l15
            const float d = cs - 2.0f * acc[v];
            if (d < bestD[v]) { bestD[v] = d; bestI[v] = n; }
        }
    }

    // reduce across the 16 lanes of each half-wave (xor masks stay in-half)
#pragma unroll
    for (int v = 0; v < 8; ++v) {
        float d = bestD[v];
        int   i = bestI[v];
#pragma unroll
        for (int m = 8; m >= 1; m >>= 1) {
            const float od = __shfl_xor(d, m, 32);
            const int   oi = __shfl_xor(i, m, 32);
            if (od < d || (od == d && oi < i)) { d = od; i = oi; }
        }
        if (l15 == 0) {                               // lane 0 -> M=v, lane 16 -> M=v+8
            const int row = pRowBase + v + 8 * half;
            partD[row * V_CHUNKS + blockIdx.y] = d;
            partI[row * V_CHUNKS + blockIdx.y] = i;
        }
    }
}

// -------------------------------------------- final reduce + embedding gather
__global__ __launch_bounds__(256) void vq_reduce_gather_kernel(const float* __restrict__ clip,
                                                               const float* __restrict__ partD,
                                                               const int*   __restrict__ partI,
                                                               float* __restrict__ outEmb,
                                                               int*   __restrict__ outIds) {
    __shared__ int s_idx;
    const int row = blockIdx.x;
    const int tid = threadIdx.x;

    if (tid < 32) {
        float d = 3.4e38f;
        int   i = 0;
        if (tid < V_CHUNKS) {
            d = partD[row * V_CHUNKS + tid];
            i = partI[row * V_CHUNKS + tid];
        }
#pragma unroll
        for (int m = 8; m >= 1; m >>= 1) {
            const float od = __shfl_xor(d, m, 32);
            const int   oi = __shfl_xor(i, m, 32);
            if (od < d || (od == d && oi < i)) { d = od; i = oi; }
        }
        if (tid == 0) { s_idx = i; outIds[row] = i; }
    }
    __syncthreads();

    const int idx = s_idx;
    // out = stop_grad(clip[idx] - p) + p == clip[idx] numerically
    for (int c = tid * 4; c < DIM; c += 256 * 4) {
        *(float4*)(outEmb + (size_t)row * DIM + c) =
            *(const float4*)(clip + (size_t)idx * DIM + c);
    }
}

// ---------------------------------------------------------------------------
extern "C" void kernel_launch(void* const* d_in, const int* in_sizes, int n_in,
                              void* d_out, int out_size, void* d_ws, size_t ws_size,
                              hipStream_t stream) {
    (void)in_sizes; (void)n_in; (void)out_size; (void)ws_size;

    const float* prompt = (const float*)d_in[0];   // [2048, 768] f32
    const float* clip   = (const float*)d_in[1];   // [49408, 768] f32

    // workspace layout
    float* csq   = (float*)d_ws;                               // VOCAB floats
    float* partD = csq + VOCAB;                                // 2048*16 floats
    int*   partI = (int*)(partD + PROMPT_LEN * V_CHUNKS);      // 2048*16 ints

    float* outEmb = (float*)d_out;                                       // [2048,768]
    int*   outIds = (int*)((float*)d_out + (size_t)PROMPT_LEN * DIM);    // [2048] int32 bits

    vq_csq_kernel<<<VOCAB / 8, 256, 0, stream>>>(clip, csq);
    vq_nn_kernel<<<dim3(P_BLOCKS, V_CHUNKS), 256, 0, stream>>>(prompt, clip, csq,
                                                               partD, partI);
    vq_reduce_gather_kernel<<<PROMPT_LEN, 256, 0, stream>>>(clip, partD, partI,
                                                            outEmb, outIds);
}